// DeformableConvolution1d_2430951490071
// MI455X (gfx1250) — compile-verified
//
#include <hip/hip_runtime.h>

typedef __attribute__((ext_vector_type(2))) float v2f;
typedef __attribute__((ext_vector_type(8))) float v8f;

#define K_TAP   3
#define C_IN    128
#define C_OUT   128
#define L_IN    4096
#define L_OUT   4094
#define CKDIM   (C_IN * K_TAP)   // 384 = GEMM K dimension, q = c*3+k (weight-natural order)
#define T_TILE  32               // output positions per workgroup
#define XS_STR  388              // t-major row stride (floats): even (8B align) and 4n+{0..3} banks
#define N_TILES (T_TILE / 16)    // 2 N-tiles of 16 per wave

__launch_bounds__(256, 1)
__global__ void deform_conv1d_wmma(const float* __restrict__ x,
                                   const float* __restrict__ offsets,
                                   const float* __restrict__ mask,
                                   const float* __restrict__ weight,
                                   const float* __restrict__ bias,
                                   float* __restrict__ out)
{
    __shared__ float xs[T_TILE * XS_STR];     // t-major: xs[t][q], 32*388*4 = 49664 B
    __shared__ int   uIdx[T_TILE * K_TAP];    // per-(t,k) left sample index
    __shared__ float g0m [T_TILE * K_TAP];    // G0 * mask
    __shared__ float g1m [T_TILE * K_TAP];    // G1 * mask

    const int tid  = threadIdx.x;
    const int wgid = blockIdx.x;
    const int b    = wgid >> 7;        // 128 t-tiles per batch
    const int tile = wgid & 127;
    const int t0   = tile * T_TILE;

    const float* xb = x + (size_t)b * C_IN * L_IN;

    // ---- Phase 1: channel-independent interpolation parameters ----
    if (tid < T_TILE * K_TAP) {
        const int t  = tid % T_TILE;
        const int k  = tid / T_TILE;
        const int tg = t0 + t;
        int u; float w0, w1;
        if (tg < L_OUT) {
            const float off = offsets[((size_t)b * L_OUT + tg) * K_TAP + k];
            const float tl  = (float)tg;
            float T = tl + (float)k + off;                       // dilation==1, stride==1
            T = fminf(fmaxf(T, tl), tl + (float)(K_TAP - 1));    // receptive-field clamp
            float fU = floorf(T);
            fU = fminf(fmaxf(fU, 0.0f), (float)(L_IN - 2));
            u = (int)fU;
            const float m  = mask[((size_t)b * K_TAP + k) * L_OUT + tg];
            w0 = fmaxf(0.0f, 1.0f - fabsf(fU - T)) * m;
            w1 = fmaxf(0.0f, 1.0f - fabsf(fU + 1.0f - T)) * m;
        } else { u = 0; w0 = 0.0f; w1 = 0.0f; }
        uIdx[tid] = u; g0m[tid] = w0; g1m[tid] = w1;
    }
    __syncthreads();

    // ---- Phase 2: gather + lerp into LDS, t-major, q = c*3 + k ----
    for (int e = tid; e < CKDIM * T_TILE; e += 256) {
        const int t  = e & (T_TILE - 1);
        const unsigned ck = (unsigned)(e >> 5);   // 0..383 in c*3+k order
        const unsigned c  = ck / 3u;
        const unsigned k  = ck - c * 3u;
        const int p  = (int)k * T_TILE + t;
        const int u  = uIdx[p];
        const float x0 = xb[(size_t)c * L_IN + u];
        const float x1 = xb[(size_t)c * L_IN + u + 1];
        xs[t * XS_STR + (int)ck] = g0m[p] * x0 + g1m[p] * x1;
    }
    __syncthreads();

    // ---- Phase 3: GEMM via V_WMMA_F32_16X16X4_F32, 4 independent chains ----
    const int wave = tid >> 5;           // 8 waves -> 8 M-tiles of 16 rows
    const int lane = tid & 31;
    const int half = lane >> 4;          // lane half selects K pair (A/B) and row half (C/D)
    const int ln16 = lane & 15;
    const int oRow = wave * 16 + ln16;   // A-matrix row for this lane

    v8f accE[N_TILES];                   // even K-quads, seeded with bias
    v8f accO[N_TILES];                   // odd K-quads, seeded with zero
    {
        const int obase = wave * 16 + 8 * half;   // C/D: VGPR r holds row r (+8 for hi half)
        #pragma unroll
        for (int r = 0; r < 8; ++r) {
            const float bv = bias[obase + r];
            #pragma unroll
            for (int j = 0; j < N_TILES; ++j) { accE[j][r] = bv; accO[j][r] = 0.0f; }
        }
    }

    const float* Wrow  = weight + (size_t)oRow * CKDIM;   // contiguous along q
    const float* xsRow0 = &xs[ln16 * XS_STR];             // N-tile j adds j*16*XS_STR
    const int    qh     = 2 * half;

    for (int kk = 0; kk < CKDIM; kk += 8) {
        const v2f a0 = *(const v2f*)(Wrow + kk + qh);          // global_load_b64
        const v2f a1 = *(const v2f*)(Wrow + kk + 4 + qh);
        #pragma unroll
        for (int j = 0; j < N_TILES; ++j) {
            const float* bp = xsRow0 + j * 16 * XS_STR + kk + qh;
            const v2f b0 = *(const v2f*)(bp);                  // ds_load_b64, bank-clean
            const v2f b1 = *(const v2f*)(bp + 4);
            accE[j] = __builtin_amdgcn_wmma_f32_16x16x4_f32(
                false, a0, false, b0, (short)0, accE[j], false, false);
            accO[j] = __builtin_amdgcn_wmma_f32_16x16x4_f32(
                false, a1, false, b1, (short)0, accO[j], false, false);
        }
    }

    // ---- Phase 4: combine chains and store ----
    float* outb = out + (size_t)b * C_OUT * L_OUT;
    #pragma unroll
    for (int j = 0; j < N_TILES; ++j) {
        const int t = t0 + j * 16 + ln16;
        if (t < L_OUT) {
            #pragma unroll
            for (int r = 0; r < 8; ++r) {
                const int o = wave * 16 + 8 * half + r;
                outb[(size_t)o * L_OUT + t] = accE[j][r] + accO[j][r];
            }
        }
    }
}

extern "C" void kernel_launch(void* const* d_in, const int* in_sizes, int n_in,
                              void* d_out, int out_size, void* d_ws, size_t ws_size,
                              hipStream_t stream) {
    const float* x       = (const float*)d_in[0];
    const float* offsets = (const float*)d_in[1];
    const float* mask    = (const float*)d_in[2];
    const float* weight  = (const float*)d_in[3];
    const float* bias    = (const float*)d_in[4];
    float* out = (float*)d_out;

    const int B = 16;
    const int tiles_per_batch = (L_OUT + T_TILE - 1) / T_TILE;  // 128
    dim3 grid(B * tiles_per_batch);
    dim3 block(256);
    deform_conv1d_wmma<<<grid, block, 0, stream>>>(x, offsets, mask, weight, bias, out);
}